// MultiClassDiceLoss_78065325572508
// MI455X (gfx1250) — compile-verified
//
#include <hip/hip_runtime.h>

// MultiClassDiceLoss on gfx1250 (MI455X).
// inputs, targets: [32, 8, 256, 256] fp32.  Output: scalar fp32.
// Memory-bound: 128 MiB streamed once -> design for bandwidth; WMMA f32
// (16x16x4, exact fp32 math) used for deterministic reductions.

typedef __attribute__((ext_vector_type(2))) float v2f;
typedef __attribute__((ext_vector_type(8))) float v8f;

#define EPS_F    1e-6f
#define NBATCH   32
#define NCLASS   8
#define SPATIAL  65536                 // 256*256
#define BSTRIDE  (NCLASS * SPATIAL)    // 524288 floats per batch
#define NCOLS    (NCLASS * SPATIAL)    // 524288 dice columns
#define SPB      64                    // spatial sites per block
#define NBLOCKS  (SPATIAL / SPB)       // 1024
#define NTHREADS 256                   // 8 wave32
#define NSLICE   4                     // batch split per block
#define BPS      (NBATCH / NSLICE)     // 8 batches per slice

// Sum a wave32's worth of 64 values per chunk with V_WMMA_F32_16X16X4_F32.
// A = all-ones 16x4 => D[m][n] = sum_k B[k][n] (layout-invariant total).
__device__ __forceinline__ v8f wmma_colsum_acc(v2f bv, v8f acc) {
  v2f ones = {1.0f, 1.0f};
  return __builtin_amdgcn_wmma_f32_16x16x4_f32(
      /*neg_a=*/false, ones, /*neg_b=*/false, bv,
      /*c_mod=*/(short)0, acc, /*reuse_a=*/false, /*reuse_b=*/false);
}

__device__ __forceinline__ float wave_total_from_wmma(v8f acc) {
  // VGPR0 of D holds colsum_n in lanes 0..15 (M=0) and again in lanes
  // 16..31 (M=8, rows identical since A is ones) -> lane-sum = 2 * total.
  float v = acc[0];
  #pragma unroll
  for (int off = 16; off >= 1; off >>= 1) v += __shfl_xor(v, off, 32);
  return 0.5f * v;
}

__global__ __launch_bounds__(NTHREADS)
void dice_main(const float* __restrict__ in, const float* __restrict__ tgt,
               float* __restrict__ blocksum) {
  __shared__ float s_pt[NSLICE][NCLASS][SPB];  // sum_b p*t  per slice
  __shared__ float s_d [NSLICE][NCLASS][SPB];  // sum_b p+t  per slice
  __shared__ float sred[NTHREADS];

  const int tid   = threadIdx.x;
  const int sl    = tid & (SPB - 1);   // spatial site within block
  const int slice = tid >> 6;          // batch slice 0..3
  const int s     = blockIdx.x * SPB + sl;
  const int b0    = slice * BPS;

  float acc_pt[NCLASS], acc_d[NCLASS];
  #pragma unroll
  for (int c = 0; c < NCLASS; ++c) { acc_pt[c] = 0.0f; acc_d[c] = 0.0f; }

  for (int b = b0; b < b0 + BPS; ++b) {
    const float* xp = in  + (size_t)b * BSTRIDE + s;
    const float* tp = tgt + (size_t)b * BSTRIDE + s;

    float x[NCLASS];
    #pragma unroll
    for (int c = 0; c < NCLASS; ++c)         // streaming: NT loads (no reuse)
      x[c] = __builtin_nontemporal_load(xp + (size_t)c * SPATIAL);

    float m = x[0];
    #pragma unroll
    for (int c = 1; c < NCLASS; ++c) m = fmaxf(m, x[c]);

    float se = 0.0f;
    #pragma unroll
    for (int c = 0; c < NCLASS; ++c) { x[c] = __expf(x[c] - m); se += x[c]; }
    const float inv = 1.0f / se;

    #pragma unroll
    for (int c = 0; c < NCLASS; ++c) {
      float t = __builtin_nontemporal_load(tp + (size_t)c * SPATIAL);
      float p = x[c] * inv;
      acc_pt[c] = fmaf(p, t, acc_pt[c]);
      acc_d[c] += p + t;
    }
  }

  #pragma unroll
  for (int c = 0; c < NCLASS; ++c) {
    s_pt[slice][c][sl] = acc_pt[c];
    s_d [slice][c][sl] = acc_d[c];
  }
  __syncthreads();

  // Merge the 4 batch slices and form dice terms for the block's 512 columns.
  float partial = 0.0f;
  #pragma unroll
  for (int k = 0; k < (NCLASS * SPB) / NTHREADS; ++k) {   // 2 columns/thread
    int col = tid + k * NTHREADS;                         // 0..511
    int c = col >> 6, q = col & (SPB - 1);
    float pt = s_pt[0][c][q] + s_pt[1][c][q] + s_pt[2][c][q] + s_pt[3][c][q];
    float dd = s_d [0][c][q] + s_d [1][c][q] + s_d [2][c][q] + s_d [3][c][q];
    partial += (2.0f * pt + EPS_F) / (dd + EPS_F);
  }
  sred[tid] = partial;
  __syncthreads();

  // Block reduction of 256 partials: wave 0, 4 accumulating f32 WMMAs.
  if (tid < 32) {                      // uniform per wave -> EXEC all-1s inside
    v8f acc = {};
    #pragma unroll
    for (int q = 0; q < NTHREADS / 64; ++q) {
      v2f bv;
      bv.x = sred[q * 64 + tid];
      bv.y = sred[q * 64 + 32 + tid];
      acc = wmma_colsum_acc(bv, acc);
    }
    float total = wave_total_from_wmma(acc);
    if (tid == 0) blocksum[blockIdx.x] = total;
  }
}

__global__ __launch_bounds__(32)
void dice_final(const float* __restrict__ blocksum, float* __restrict__ out) {
  const int lane = threadIdx.x;        // single wave32, EXEC all-1s
  v8f acc = {};
  #pragma unroll
  for (int q = 0; q < NBLOCKS / 64; ++q) {   // 16 accumulating WMMAs
    v2f bv;
    bv.x = blocksum[q * 64 + lane];
    bv.y = blocksum[q * 64 + 32 + lane];
    acc = wmma_colsum_acc(bv, acc);
  }
  float total = wave_total_from_wmma(acc);
  if (lane == 0) out[0] = 1.0f - total / (float)NCOLS;
}

extern "C" void kernel_launch(void* const* d_in, const int* in_sizes, int n_in,
                              void* d_out, int out_size, void* d_ws, size_t ws_size,
                              hipStream_t stream) {
  (void)in_sizes; (void)n_in; (void)out_size; (void)ws_size;
  const float* in  = (const float*)d_in[0];
  const float* tgt = (const float*)d_in[1];
  float* out = (float*)d_out;
  float* bs  = (float*)d_ws;           // needs NBLOCKS * 4 = 4 KiB scratch

  dice_main <<<NBLOCKS, NTHREADS, 0, stream>>>(in, tgt, bs);
  dice_final<<<1, 32, 0, stream>>>(bs, out);
}